// TopK_82746839925404
// MI455X (gfx1250) — compile-verified
//
#include <hip/hip_runtime.h>
#include <stdint.h>

// Problem constants from the reference: x is [rows, 16384] fp32, k is a scalar.
#define COLS 16384
#define NTHREADS 256
#define CHUNKS_PER_THREAD (COLS / (NTHREADS * 4))  // 16 float4 chunks per thread
#define CAND_CAP 2048                              // candidate-list capacity (keys)

typedef float f32x4 __attribute__((ext_vector_type(4)));

// Monotonic bijection float -> uint32 (order-preserving: bigger float = bigger key).
__device__ __forceinline__ uint32_t f2key(float f) {
    uint32_t u = __float_as_uint(f);
    return (u & 0x80000000u) ? ~u : (u | 0x80000000u);
}

__device__ __forceinline__ float gate_val(float f, uint32_t T, uint32_t r,
                                          uint32_t* tie_counter) {
    const uint32_t key = f2key(f);
    if (key > T) return f;
    if (key == T) {
        // Exact-threshold ties: admit exactly r of them (ticket via LDS atomic).
        const uint32_t ticket = atomicAdd(tie_counter, 1u);
        return (ticket < r) ? f : 0.0f;
    }
    return 0.0f;
}

__global__ __launch_bounds__(NTHREADS) void topk_mask_kernel(
    const float* __restrict__ x, const int* __restrict__ kptr,
    float* __restrict__ out, int rows)
{
    __shared__ __align__(16) float s_row[COLS];   // 64 KB staged row
    __shared__ uint32_t s_hist[256];              // radix histogram
    __shared__ uint32_t s_scan[256];              // suffix-sum workspace
    __shared__ uint32_t s_cand[CAND_CAP];         // compacted candidate keys (8 KB)
    __shared__ uint32_t s_sel[4];  // [0]=sel bin, [1]=remaining, [2]=tie ticket, [3]=cand count

    const int row = blockIdx.x;
    if (row >= rows) return;
    const int tid = threadIdx.x;

    const uint64_t gbase = (uint64_t)(uintptr_t)(x + (size_t)row * COLS);
    // Low 32 bits of a generic pointer to LDS == wave-relative LDS byte offset.
    const uint32_t lbase = (uint32_t)(uintptr_t)(void*)&s_row[0];

    // ---- Stage the row HBM -> LDS with CDNA5 async copies (ASYNCcnt-tracked) ----
    #pragma unroll
    for (int i = 0; i < CHUNKS_PER_THREAD; ++i) {
        const uint32_t boff  = (uint32_t)(tid + i * NTHREADS) * 16u;  // byte offset in row
        const uint32_t laddr = lbase + boff;                          // LDS byte address
        // GVS mode: vdst = LDS addr VGPR, vaddr = 32-bit offset VGPR, saddr = 64-bit base.
        asm volatile("global_load_async_to_lds_b128 %0, %1, %2"
                     :: "v"(laddr), "v"(boff), "s"(gbase) : "memory");
    }

    const uint32_t kk = (uint32_t)kptr[0];
    f32x4* orow4 = (f32x4*)(out + (size_t)row * COLS);

    if (kk == 0u) {   // degenerate: gate is all zeros
#if __has_builtin(__builtin_amdgcn_s_wait_asynccnt)
        __builtin_amdgcn_s_wait_asynccnt(0);
#else
        asm volatile("s_wait_asynccnt 0x0" ::: "memory");
#endif
        const f32x4 z = {0.f, 0.f, 0.f, 0.f};
        #pragma unroll
        for (int i = 0; i < CHUNKS_PER_THREAD; ++i)
            __builtin_nontemporal_store(z, orow4 + (tid + i * NTHREADS));
        return;
    }

    if (tid == 0) { s_sel[1] = kk; s_sel[2] = 0u; s_sel[3] = 0u; }

#if __has_builtin(__builtin_amdgcn_s_wait_asynccnt)
    __builtin_amdgcn_s_wait_asynccnt(0);
#else
    asm volatile("s_wait_asynccnt 0x0" ::: "memory");
#endif
    __syncthreads();  // everyone's async chunks visible block-wide; s_sel init visible

    // Parallel bin selection: suffix-sum s_hist into s_scan, pick bin where
    // S[t] >= rem > S[t+1]; update s_sel[0]=bin, s_sel[1]=rem-S[t+1].
    auto select_bin = [&]() {
        s_scan[tid] = s_hist[tid];
        __syncthreads();
        for (int off = 1; off < 256; off <<= 1) {
            uint32_t v = s_scan[tid];
            if (tid + off < 256) v += s_scan[tid + off];
            __syncthreads();
            s_scan[tid] = v;
            __syncthreads();
        }
        const uint32_t S_t  = s_scan[tid];
        const uint32_t S_t1 = (tid < 255) ? s_scan[tid + 1] : 0u;
        const uint32_t rem  = s_sel[1];
        __syncthreads();
        if (S_t >= rem && S_t1 < rem) { s_sel[0] = (uint32_t)tid; s_sel[1] = rem - S_t1; }
        __syncthreads();
    };

    // ---- Pass 1: histogram of the key MSB over the full row ----
    s_hist[tid] = 0u;
    __syncthreads();
    for (int i = 0; i < CHUNKS_PER_THREAD; ++i) {
        const float4 v = reinterpret_cast<const float4*>(s_row)[tid + i * NTHREADS];
        atomicAdd(&s_hist[f2key(v.x) >> 24], 1u);
        atomicAdd(&s_hist[f2key(v.y) >> 24], 1u);
        atomicAdd(&s_hist[f2key(v.z) >> 24], 1u);
        atomicAdd(&s_hist[f2key(v.w) >> 24], 1u);
    }
    __syncthreads();
    select_bin();

    uint32_t selb   = s_sel[0];
    uint32_t prefix = selb << 24;
    uint32_t pmask  = 0xFF000000u;

    // ---- Compact surviving keys (top byte == selb) into s_cand if they fit ----
    const uint32_t surv = s_hist[selb];            // still valid (scan used s_scan)
    const bool use_cand = (surv <= CAND_CAP);
    if (use_cand) {
        for (int i = 0; i < CHUNKS_PER_THREAD; ++i) {
            const float4 v = reinterpret_cast<const float4*>(s_row)[tid + i * NTHREADS];
            const uint32_t key[4] = {f2key(v.x), f2key(v.y), f2key(v.z), f2key(v.w)};
            #pragma unroll
            for (int j = 0; j < 4; ++j) {
                if ((key[j] >> 24) == selb) {
                    const uint32_t pos = atomicAdd(&s_sel[3], 1u);
                    s_cand[pos] = key[j];
                }
            }
        }
    }
    __syncthreads();
    const uint32_t cand_n = s_sel[3];

    // ---- Passes 2..4: refine over candidates (or full row fallback) ----
    for (int byte = 2; byte >= 0; --byte) {
        s_hist[tid] = 0u;
        __syncthreads();
        const uint32_t shift = (uint32_t)byte * 8u;
        if (use_cand) {
            for (uint32_t idx = (uint32_t)tid; idx < cand_n; idx += NTHREADS) {
                const uint32_t key = s_cand[idx];
                if ((key & pmask) == prefix)
                    atomicAdd(&s_hist[(key >> shift) & 0xFFu], 1u);
            }
        } else {
            for (int i = 0; i < CHUNKS_PER_THREAD; ++i) {
                const float4 v = reinterpret_cast<const float4*>(s_row)[tid + i * NTHREADS];
                const uint32_t key[4] = {f2key(v.x), f2key(v.y), f2key(v.z), f2key(v.w)};
                #pragma unroll
                for (int j = 0; j < 4; ++j)
                    if ((key[j] & pmask) == prefix)
                        atomicAdd(&s_hist[(key[j] >> shift) & 0xFFu], 1u);
            }
        }
        __syncthreads();
        select_bin();
        prefix |= (s_sel[0] << shift);
        pmask  |= (0xFFu << shift);
    }

    // ---- Emit gated output straight from LDS (write-once, NT b128 stores) ----
    const uint32_t T = prefix;     // key of the k-th largest element
    const uint32_t r = s_sel[1];   // number of exact-threshold elements to keep
    #pragma unroll
    for (int i = 0; i < CHUNKS_PER_THREAD; ++i) {
        const int c4 = tid + i * NTHREADS;
        const float4 v = reinterpret_cast<const float4*>(s_row)[c4];
        f32x4 o;
        o.x = gate_val(v.x, T, r, &s_sel[2]);
        o.y = gate_val(v.y, T, r, &s_sel[2]);
        o.z = gate_val(v.z, T, r, &s_sel[2]);
        o.w = gate_val(v.w, T, r, &s_sel[2]);
        __builtin_nontemporal_store(o, orow4 + c4);
    }
}

extern "C" void kernel_launch(void* const* d_in, const int* in_sizes, int n_in,
                              void* d_out, int out_size, void* d_ws, size_t ws_size,
                              hipStream_t stream) {
    (void)n_in; (void)out_size; (void)d_ws; (void)ws_size;
    const float* x    = (const float*)d_in[0];
    const int*   kptr = (const int*)d_in[1];
    float*       out  = (float*)d_out;
    const int rows = in_sizes[0] / COLS;   // 4096 for the reference shapes
    topk_mask_kernel<<<rows, NTHREADS, 0, stream>>>(x, kptr, out, rows);
}